// Self_L3_41944650612977
// MI455X (gfx1250) — compile-verified
//
#include <hip/hip_runtime.h>
#include <math.h>

typedef __attribute__((ext_vector_type(2))) float v2f;
typedef __attribute__((ext_vector_type(8))) float v8f;

#define B_ROWS 8192
#define TILE 16
#define NTILES (B_ROWS / TILE)          // 512 tiles per dimension
#define JCHUNK 64                        // j-tiles handled per wave
#define NJCHUNKS (NTILES / JCHUNK)       // 8
#define WAVES_PER_BLOCK 8
#define TOTAL_WAVES (NTILES * NJCHUNKS)  // 4096
#define P2_BLOCKS (TOTAL_WAVES / WAVES_PER_BLOCK) // 512

// Workspace layout (bytes):
//   LP  : float2[B_ROWS]       @ 0          (64 KB)  {logp0, logp1} per row
//   AD  : float2[B_ROWS]       @ 64 KB      (64 KB)  negated D = q - p per row
//   AB  : float [B_ROWS]       @ 128 KB     (32 KB)  bias A_i per row
//   PART: float [TOTAL_WAVES]  @ 160 KB     (16 KB)  per-wave partial sums

__global__ void prep_kernel(const float* __restrict__ s,
                            const float* __restrict__ cw,
                            const long long* __restrict__ t,
                            float2* __restrict__ LP,
                            float2* __restrict__ AD,
                            float* __restrict__ AB) {
    int i = blockIdx.x * blockDim.x + threadIdx.x;
    if (i >= B_ROWS) return;
    float s0 = s[2 * i + 0];
    float s1 = s[2 * i + 1];
    float m  = fmaxf(s0, s1);
    float e0 = __expf(s0 - m), e1 = __expf(s1 - m);
    float lse = m + __logf(e0 + e1);
    float lp0 = s0 - lse, lp1 = s1 - lse;
    float p0 = __expf(lp0), p1 = __expf(lp1);
    long long ti = t[i];
    float q0 = (ti == 0) ? 1.0f : 1.0e-4f;   // clip(one_hot, 1e-4, 1)
    float q1 = (ti == 1) ? 1.0f : 1.0e-4f;
    q0 *= cw[0];
    q1 *= cw[1];
    float A = p0 * lp0 + p1 * lp1 - q0 * __logf(q0) - q1 * __logf(q1);
    LP[i] = make_float2(lp0, lp1);
    AD[i] = make_float2(q0 - p0, q1 - p1);   // negated D so wmma computes M directly
    AB[i] = A;
}

// One wave handles one 16-row i-tile over a 64-tile j-chunk.
// M_tile = (-D_tile) x logp_tile^T + bias  via v_wmma_f32_16x16x4_f32 (K=2 live).
__global__ void mtile_kernel(const float2* __restrict__ LP,
                             const float2* __restrict__ AD,
                             const float* __restrict__ AB,
                             float* __restrict__ PART) {
    const int lane = threadIdx.x & 31;
    const int wave = threadIdx.x >> 5;
    const int wgid = blockIdx.x * WAVES_PER_BLOCK + wave;   // 0..4095
    const int it   = wgid >> 3;          // i-tile 0..511
    const int jc   = wgid & (NJCHUNKS - 1); // j-chunk 0..7
    const int half = lane >> 4;          // 0: lanes 0-15 (K0,K1) ; 1: lanes 16-31 (K2,K3)
    const int l15  = lane & 15;

    // A operand (16x4 f32): lane(0-15) row=l15 holds {K0,K1}; lanes 16-31 hold K2,K3 = 0.
    float2 adv = AD[it * TILE + l15];
    v2f a;
    a.x = half ? 0.0f : adv.x;
    a.y = half ? 0.0f : adv.y;

    // C operand (16x16 f32): VGPR r, lane n(0-15) -> M = r + 8*half. Bias = A_i broadcast over columns.
    v8f c;
#pragma unroll
    for (int r = 0; r < 8; ++r)
        c[r] = AB[it * TILE + r + half * 8];

    v8f acc = {0.f, 0.f, 0.f, 0.f, 0.f, 0.f, 0.f, 0.f};
    const int j0 = jc * JCHUNK;
    for (int jt = j0; jt < j0 + JCHUNK; ++jt) {
        // B operand (4x16 f32): lanes 0-15 col=l15 hold rows K0,K1; lanes 16-31 -> K2,K3 = 0.
        float2 lpv = LP[jt * TILE + l15];
        v2f b;
        b.x = half ? 0.0f : lpv.x;
        b.y = half ? 0.0f : lpv.y;
        // 8 args: (neg_a, A, neg_b, B, c_mod, C, reuse_a, reuse_b)
        v8f d = __builtin_amdgcn_wmma_f32_16x16x4_f32(
            false, a, false, b, (short)0, c, false, false);
#pragma unroll
        for (int r = 0; r < 8; ++r)
            acc[r] += fabsf(d[r]);
    }

    float sum = 0.f;
#pragma unroll
    for (int r = 0; r < 8; ++r) sum += acc[r];
    // wave32 butterfly reduction
    for (int off = 16; off > 0; off >>= 1)
        sum += __shfl_xor(sum, off, 32);
    if (lane == 0) PART[wgid] = sum;
}

__global__ void finalize_kernel(const float* __restrict__ PART,
                                float* __restrict__ out) {
    __shared__ float sm[1024];
    int tid = threadIdx.x;
    float v = 0.f;
#pragma unroll
    for (int k = 0; k < TOTAL_WAVES / 1024; ++k)
        v += PART[tid + k * 1024];
    sm[tid] = v;
    __syncthreads();
    for (int sft = 512; sft > 0; sft >>= 1) {
        if (tid < sft) sm[tid] += sm[tid + sft];
        __syncthreads();
    }
    if (tid == 0)
        out[0] = (1.0e-4f + sm[0]) / ((float)B_ROWS * (float)B_ROWS);
}

extern "C" void kernel_launch(void* const* d_in, const int* in_sizes, int n_in,
                              void* d_out, int out_size, void* d_ws, size_t ws_size,
                              hipStream_t stream) {
    const float*     s  = (const float*)d_in[0];
    const float*     cw = (const float*)d_in[1];
    const long long* t  = (const long long*)d_in[2];
    float* out = (float*)d_out;

    char* ws = (char*)d_ws;
    float2* LP   = (float2*)(ws);
    float2* AD   = (float2*)(ws + (size_t)B_ROWS * 8);
    float*  AB   = (float*) (ws + (size_t)B_ROWS * 16);
    float*  PART = (float*) (ws + (size_t)B_ROWS * 20);

    prep_kernel<<<B_ROWS / 256, 256, 0, stream>>>(s, cw, t, LP, AD, AB);
    mtile_kernel<<<P2_BLOCKS, WAVES_PER_BLOCK * 32, 0, stream>>>(LP, AD, AB, PART);
    finalize_kernel<<<1, 1024, 0, stream>>>(PART, out);
}